// GroupingModel_23270132809845
// MI455X (gfx1250) — compile-verified
//
#include <hip/hip_runtime.h>

// ---------------------------------------------------------------------------
// Shapes (compile-time constants from the reference)
// ---------------------------------------------------------------------------
constexpr int cB = 8, cQ = 256, cK = 2048, cD = 256, cN = 17, cC = 32, cH = 544;

// d_out layout: outputs concatenated flat in return order (float32)
constexpr long long LOC_OFF  = 0;                       // [B,Q,N,2]  69632
constexpr long long PER_OFF  = 69632;                   // [B,Q,1]     2048
constexpr long long HLOC_OFF = 71680;                   // [B,Q,N,2]  69632
constexpr long long VIS_OFF  = 141312;                  // [B,Q,N,1]  34816
constexpr long long ATTN_OFF = 176128;                  // [B,Q,N,K]  71303168
constexpr long long WRAW_OFF = 71479296;                // [B,Q,N,K]  71303168

typedef float v2f __attribute__((ext_vector_type(2)));
typedef float v8f __attribute__((ext_vector_type(8)));

// Full-precision f32 WMMA: D(16x16,f32) = A(16x4,f32) x B(4x16,f32) + C
static __device__ __forceinline__ v8f wmma4(v2f a, v2f b, v8f c) {
  return __builtin_amdgcn_wmma_f32_16x16x4_f32(
      /*neg_a=*/false, a, /*neg_b=*/false, b,
      /*c_mod=*/(short)0, c, /*reuse_a=*/false, /*reuse_b=*/false);
}

// ---------------------------------------------------------------------------
// mask sanitize: flags[b] = 1 iff all 2048 mask entries of batch b are set
// ---------------------------------------------------------------------------
__global__ __launch_bounds__(256) void mask_flags_kernel(const int* __restrict__ mask,
                                                         int* __restrict__ flags) {
  int b = blockIdx.x, tid = threadIdx.x;
  int all = 1;
#pragma unroll
  for (int i = 0; i < 8; ++i) all &= (mask[b * cK + tid + i * 256] != 0) ? 1 : 0;
  for (int off = 16; off > 0; off >>= 1) all &= __shfl_xor(all, off);
  __shared__ int red[8];
  if ((tid & 31) == 0) red[tid >> 5] = all;
  __syncthreads();
  if (tid == 0) {
    int a = 1;
#pragma unroll
    for (int j = 0; j < 8; ++j) a &= red[j];
    flags[b] = a;
  }
}

// ---------------------------------------------------------------------------
// out[M,Hn] = (X1 (+X2)) [M,Dk] @ W[Hn,Dk]^T + bias ;  one 16x16 tile / wave
// HAS_X2 templated so the embed-add variant has straight-line loads (no exec
// mask save/restore per k-step).
// ---------------------------------------------------------------------------
template <bool HAS_X2>
__global__ __launch_bounds__(256) void proj_gemm_kernel(
    const float* __restrict__ X1, const float* __restrict__ X2,
    const float* __restrict__ W, const float* __restrict__ bias,
    float* __restrict__ out, int Dk, int Hn, int nTiles, int hTiles) {
  int wave = threadIdx.x >> 5, lane = threadIdx.x & 31;
  int tile = blockIdx.x * 8 + wave;
  if (tile >= nTiles) return;
  int mt = tile / hTiles, ht = tile % hTiles;
  int m0 = mt * 16, h0 = ht * 16;
  int l15 = lane & 15, kh = (lane >> 4) * 2;
  const float* arow = X1 + (size_t)(m0 + l15) * Dk + kh;
  const float* arow2 = HAS_X2 ? X2 + (size_t)(m0 + l15) * Dk + kh : nullptr;
  const float* brow = W + (size_t)(h0 + l15) * Dk + kh;
  v8f acc = {};
#pragma unroll 4
  for (int k0 = 0; k0 < Dk; k0 += 4) {
    v2f a = *(const v2f*)(arow + k0);
    if (HAS_X2) {
      v2f a2 = *(const v2f*)(arow2 + k0);
      a.x += a2.x;
      a.y += a2.y;
    }
    v2f b = *(const v2f*)(brow + k0);
    acc = wmma4(a, b, acc);
  }
  float bv = bias[h0 + l15];
#pragma unroll
  for (int v = 0; v < 8; ++v) {
    int row = m0 + v + (lane >> 4) * 8;
    out[(size_t)row * Hn + h0 + l15] = acc[v] + bv;
  }
}

// ---------------------------------------------------------------------------
// weights_raw[b,q,n,k] = scale * <q[b,q,n,:], k[b,k,n,:]>  (masked -> -inf)
// per (b,n): [256 x 32] x [32 x 2048] GEMM, 8 WMMA steps per tile
// ---------------------------------------------------------------------------
__global__ __launch_bounds__(256) void scores_kernel(
    const float* __restrict__ qbuf, const float* __restrict__ kbuf,
    const int* __restrict__ mask, const int* __restrict__ flags,
    float* __restrict__ wraw, int nTiles) {
  int wave = threadIdx.x >> 5, lane = threadIdx.x & 31;
  int tile = blockIdx.x * 8 + wave;
  if (tile >= nTiles) return;
  int bn = tile >> 11;           // 2048 tiles per (b,n)
  int rem = tile & 2047;
  int qt = rem >> 7, kt = rem & 127;
  int b = bn / cN, n = bn % cN;
  int q0 = qt * 16, k0 = kt * 16;
  int l15 = lane & 15, kh = (lane >> 4) * 2;
  const float* qrow = qbuf + (size_t)(b * cQ + q0 + l15) * cH + n * cC + kh;
  const float* krow = kbuf + (size_t)(b * cK + k0 + l15) * cH + n * cC + kh;
  v8f acc = {};
#pragma unroll
  for (int c0 = 0; c0 < cC; c0 += 4) {
    v2f a = *(const v2f*)(qrow + c0);
    v2f bb = *(const v2f*)(krow + c0);
    acc = wmma4(a, bb, acc);
  }
  const float scale = 0.17677669529663687f;  // 1/sqrt(32)
  int kk = k0 + l15;
  int msk = flags[b] ? 0 : mask[b * cK + kk];  // sanitize: all-masked -> none
  float neginf = -__builtin_inff();
#pragma unroll
  for (int v = 0; v < 8; ++v) {
    int q = q0 + v + (lane >> 4) * 8;
    size_t idx = ((size_t)(b * cQ + q) * cN + n) * cK + kk;
    wraw[idx] = msk ? neginf : acc[v] * scale;
  }
}

// ---------------------------------------------------------------------------
// Fused per-row softmax over K=2048: writes attn, loc_pred, hard_loc_pred
// one block per (b,q,n) row; 256 threads x 8 elements
// ---------------------------------------------------------------------------
__global__ __launch_bounds__(256) void softmax_kernel(
    const float* __restrict__ wraw, const float* __restrict__ coords,
    float* __restrict__ attn, float* __restrict__ loc, float* __restrict__ hloc) {
  int r = blockIdx.x, tid = threadIdx.x;
  int lane = tid & 31, wave = tid >> 5;
  int b = r / (cQ * cN);
  size_t base = (size_t)r * cK;

  float w[8], kx[8], ky[8];
#pragma unroll
  for (int i = 0; i < 8; ++i) {
    int k = tid + i * 256;
    w[i] = wraw[base + k];
    v2f cp = *(const v2f*)(coords + ((size_t)b * cK + k) * 2);
    kx[i] = cp.x;
    ky[i] = cp.y;
  }
  // block max
  float m = w[0];
#pragma unroll
  for (int i = 1; i < 8; ++i) m = fmaxf(m, w[i]);
  for (int off = 16; off > 0; off >>= 1) m = fmaxf(m, __shfl_xor(m, off));
  __shared__ float redM[8];
  if (lane == 0) redM[wave] = m;
  __syncthreads();
  float M = redM[0];
#pragma unroll
  for (int j = 1; j < 8; ++j) M = fmaxf(M, redM[j]);

  float e[8], s = 0.f, cnt = 0.f, cx = 0.f, cy = 0.f, lx = 0.f, ly = 0.f;
#pragma unroll
  for (int i = 0; i < 8; ++i) {
    float ei = __expf(w[i] - M);  // exp(-inf - M) = 0 for masked entries
    e[i] = ei;
    s += ei;
    lx += ei * kx[i];
    ly += ei * ky[i];
    if (w[i] == M) { cnt += 1.f; cx += kx[i]; cy += ky[i]; }
  }
  float vals[6] = {s, cnt, cx, cy, lx, ly};
  __shared__ float red6[8][6];
#pragma unroll
  for (int j = 0; j < 6; ++j) {
    float t = vals[j];
    for (int off = 16; off > 0; off >>= 1) t += __shfl_xor(t, off);
    if (lane == 0) red6[wave][j] = t;
  }
  __syncthreads();
  float S = 0, CNT = 0, CX = 0, CY = 0, LX = 0, LY = 0;
#pragma unroll
  for (int wv = 0; wv < 8; ++wv) {
    S += red6[wv][0]; CNT += red6[wv][1]; CX += red6[wv][2];
    CY += red6[wv][3]; LX += red6[wv][4]; LY += red6[wv][5];
  }
  float inv = 1.0f / S;
#pragma unroll
  for (int i = 0; i < 8; ++i) attn[base + tid + i * 256] = e[i] * inv;
  if (tid == 0) {
    loc[(size_t)r * 2 + 0] = LX * inv;
    loc[(size_t)r * 2 + 1] = LY * inv;
    float ic = 1.0f / CNT;
    hloc[(size_t)r * 2 + 0] = CX * ic;
    hloc[(size_t)r * 2 + 1] = CY * ic;
  }
}

// ---------------------------------------------------------------------------
// result[b,q,n,c] = sum_k attn[b,q,n,k] * v[b,k,n*C+c]
// per (b,n): [256 x 2048] x [2048 x 32] GEMM, 512 WMMA steps per tile
// ---------------------------------------------------------------------------
__global__ __launch_bounds__(256) void attnv_kernel(
    const float* __restrict__ attn, const float* __restrict__ vbuf,
    float* __restrict__ resbuf, int nTiles) {
  int wave = threadIdx.x >> 5, lane = threadIdx.x & 31;
  int tile = blockIdx.x * 8 + wave;
  if (tile >= nTiles) return;
  int bn = tile >> 5;            // 32 tiles per (b,n)
  int rem = tile & 31;
  int qt = rem >> 1, ct = rem & 1;
  int b = bn / cN, n = bn % cN;
  int q0 = qt * 16, c0 = ct * 16;
  int l15 = lane & 15, kh = (lane >> 4) * 2;
  const float* arow =
      attn + ((size_t)(b * cQ + q0 + l15) * cN + n) * cK + kh;
  const float* vbase = vbuf + ((size_t)b * cK + kh) * cH + n * cC + c0 + l15;
  v8f acc = {};
#pragma unroll 4
  for (int k0 = 0; k0 < cK; k0 += 4) {
    v2f a = *(const v2f*)(arow + k0);
    v2f bb;
    bb.x = vbase[(size_t)k0 * cH];
    bb.y = vbase[(size_t)(k0 + 1) * cH];
    acc = wmma4(a, bb, acc);
  }
#pragma unroll
  for (int v = 0; v < 8; ++v) {
    int q = q0 + v + (lane >> 4) * 8;
    resbuf[((size_t)(b * cQ + q) * cN + n) * cC + c0 + l15] = acc[v];
  }
}

// ---------------------------------------------------------------------------
// MLP head: out[r] = relu(X[r,:] @ W1^T + b1) @ w2 + b2
// MODE 0: X = Xa[r, 0:256]                      (person head, Kdim=256)
// MODE 1: X = [Xa[r/N, 0:256] ; Xb[r, 0:32]]    (vis head,    Kdim=288)
// One wave owns 16 rows and loops all 16 hidden tiles (no atomics).
// Concat boundary is 4-aligned, so MODE 1 splits into two clean loops.
// ---------------------------------------------------------------------------
template <int MODE>
__global__ __launch_bounds__(256) void mlp_head_kernel(
    const float* __restrict__ Xa, const float* __restrict__ Xb,
    const float* __restrict__ W1, const float* __restrict__ b1,
    const float* __restrict__ w2, const float* __restrict__ b2,
    float* __restrict__ out, int Kdim, int nTiles) {
  int wave = threadIdx.x >> 5, lane = threadIdx.x & 31;
  int tile = blockIdx.x * 8 + wave;
  if (tile >= nTiles) return;
  int r0 = tile * 16;
  int l15 = lane & 15, kh = (lane >> 4) * 2;
  int r = r0 + l15;
  const float* arow = (MODE == 1) ? Xa + (size_t)(r / cN) * cD + kh
                                  : Xa + (size_t)r * cD + kh;
  const float* arow2 = (MODE == 1) ? Xb + (size_t)r * cC + kh : nullptr;
  float sacc[8] = {0, 0, 0, 0, 0, 0, 0, 0};
  for (int ht = 0; ht < 16; ++ht) {
    int h = ht * 16 + l15;
    const float* brow = W1 + (size_t)h * Kdim + kh;
    v8f acc = {};
#pragma unroll 4
    for (int k0 = 0; k0 < cD; k0 += 4) {  // first segment: Xa
      v2f a = *(const v2f*)(arow + k0);
      v2f bb = *(const v2f*)(brow + k0);
      acc = wmma4(a, bb, acc);
    }
    if (MODE == 1) {
#pragma unroll
      for (int k0 = cD; k0 < cD + cC; k0 += 4) {  // second segment: Xb
        v2f a = *(const v2f*)(arow2 + (k0 - cD));
        v2f bb = *(const v2f*)(brow + k0);
        acc = wmma4(a, bb, acc);
      }
    }
    float w2h = w2[h], b1h = b1[h];
#pragma unroll
    for (int v = 0; v < 8; ++v) {
      float hid = acc[v] + b1h;
      hid = hid > 0.f ? hid : 0.f;
      sacc[v] += hid * w2h;
    }
  }
  float b2v = b2[0];
#pragma unroll
  for (int v = 0; v < 8; ++v) {
    float s = sacc[v];
    for (int off = 1; off < 16; off <<= 1) s += __shfl_xor(s, off);
    if (l15 == 0) out[r0 + v + (lane >> 4) * 8] = s + b2v;
  }
}

// ---------------------------------------------------------------------------
// Launch
// ---------------------------------------------------------------------------
extern "C" void kernel_launch(void* const* d_in, const int* in_sizes, int n_in,
                              void* d_out, int out_size, void* d_ws, size_t ws_size,
                              hipStream_t stream) {
  const float* p_nodes   = (const float*)d_in[0];
  const float* kp_nodes  = (const float*)d_in[1];
  const float* kp_coords = (const float*)d_in[2];
  const float* p_embed   = (const float*)d_in[3];
  const float* kp_embed  = (const float*)d_in[4];
  const int*   mask      = (const int*)d_in[5];
  const float* q_w = (const float*)d_in[6],  *q_b = (const float*)d_in[7];
  const float* k_w = (const float*)d_in[8],  *k_b = (const float*)d_in[9];
  const float* v_w = (const float*)d_in[10], *v_b = (const float*)d_in[11];
  const float* vis_w1 = (const float*)d_in[12], *vis_b1 = (const float*)d_in[13];
  const float* vis_w2 = (const float*)d_in[14], *vis_b2 = (const float*)d_in[15];
  const float* per_w1 = (const float*)d_in[16], *per_b1 = (const float*)d_in[17];
  const float* per_w2 = (const float*)d_in[18], *per_b2 = (const float*)d_in[19];

  float* ws = (float*)d_ws;
  float* qbuf   = ws;                                   // [B,Q,H]
  float* kbuf   = qbuf + (size_t)cB * cQ * cH;          // [B,K,H]
  float* vbuf   = kbuf + (size_t)cB * cK * cH;          // [B,K,H]
  float* resbuf = vbuf + (size_t)cB * cK * cH;          // [B,Q,N,C]
  int*   flags  = (int*)(resbuf + (size_t)cB * cQ * cN * cC);

  float* out = (float*)d_out;

  mask_flags_kernel<<<cB, 256, 0, stream>>>(mask, flags);

  {  // q projection: M = B*Q (with positional embed)
    int M = cB * cQ, nT = (M / 16) * (cH / 16);
    proj_gemm_kernel<true><<<(nT + 7) / 8, 256, 0, stream>>>(
        p_nodes, p_embed, q_w, q_b, qbuf, cD, cH, nT, cH / 16);
  }
  {  // k projection: M = B*K (with positional embed)
    int M = cB * cK, nT = (M / 16) * (cH / 16);
    proj_gemm_kernel<true><<<(nT + 7) / 8, 256, 0, stream>>>(
        kp_nodes, kp_embed, k_w, k_b, kbuf, cD, cH, nT, cH / 16);
  }
  {  // v projection: M = B*K (no embed)
    int M = cB * cK, nT = (M / 16) * (cH / 16);
    proj_gemm_kernel<false><<<(nT + 7) / 8, 256, 0, stream>>>(
        kp_nodes, nullptr, v_w, v_b, vbuf, cD, cH, nT, cH / 16);
  }
  {  // scores -> weights_raw
    int nT = (cB * cN) * (cQ / 16) * (cK / 16);  // 278528
    scores_kernel<<<(nT + 7) / 8, 256, 0, stream>>>(
        qbuf, kbuf, mask, flags, out + WRAW_OFF, nT);
  }
  // softmax + loc + hard_loc (one block per row)
  softmax_kernel<<<cB * cQ * cN, 256, 0, stream>>>(
      out + WRAW_OFF, kp_coords, out + ATTN_OFF, out + LOC_OFF, out + HLOC_OFF);
  {  // attn @ v -> result
    int nT = (cB * cN) * (cQ / 16) * (cC / 16);  // 4352
    attnv_kernel<<<(nT + 7) / 8, 256, 0, stream>>>(
        out + ATTN_OFF, vbuf, resbuf, nT);
  }
  {  // vis head (MODE 1, Kdim = 288)
    int nT = (cB * cQ * cN) / 16;  // 2176
    mlp_head_kernel<1><<<(nT + 7) / 8, 256, 0, stream>>>(
        p_nodes, resbuf, vis_w1, vis_b1, vis_w2, vis_b2,
        out + VIS_OFF, cD + cC, nT);
  }
  {  // person head (MODE 0, Kdim = 256)
    int nT = (cB * cQ) / 16;  // 128
    mlp_head_kernel<0><<<(nT + 7) / 8, 256, 0, stream>>>(
        p_nodes, nullptr, per_w1, per_b1, per_w2, per_b2,
        out + PER_OFF, cD, nT);
  }
}